// RBF_83889301225989
// MI455X (gfx1250) — compile-verified
//
#include <hip/hip_runtime.h>
#include <hip/hip_bf16.h>

#define BETA   0.04f
#define DELAY  100
#define T_DIM  4096
#define M_DIM  16384

typedef __attribute__((ext_vector_type(2))) float v2f;
typedef __attribute__((ext_vector_type(8))) float v8f;

// SCL = -log2(e)/BETA :  exp(-d2/beta) == exp2(d2 * SCL)
#define SCL   (-36.06737602222409f)
#define N2SCL (72.13475204444818f)   // -2 * SCL

// One wave computes a 16x64 strip (one M-tile x 4 N-tiles) of out[t, m].
// Per N-tile: two V_WMMA_F32_16X16X4_F32 with SCL pre-folded into B:
//   SCL*d2 = [x0, x1, |x|^2, 1] . [N2SCL*c0, N2SCL*c1, SCL, SCL*|c|^2]^T
// then out = exp2(SCL*d2_x) + exp2(SCL*d2_xd).
__global__ void __launch_bounds__(256)
rbf_wmma_kernel(const float* __restrict__ sig,   // (T_DIM + DELAY, 2)
                const float* __restrict__ cen,   // (M_DIM, 2)
                float* __restrict__ out)         // (T_DIM, M_DIM)
{
    const int lane  = threadIdx.x & 31;
    const int wave  = threadIdx.x >> 5;
    const int nBase = (blockIdx.x * 8 + wave) << 6;  // 4 N-tiles = 64 columns
    const int m0    = blockIdx.y << 4;               // 16 rows

    const int  r  = lane & 15;
    const bool hi = (lane >= 16);

    // ---- A matrices (16x4), loaded/built once, reused for 4 N-tiles.
    // A layout: V0 = K0 (lanes 0-15) / K2 (lanes 16-31); V1 = K1 / K3.
    const float x0 = sig[(DELAY + m0 + r) * 2 + 0];
    const float x1 = sig[(DELAY + m0 + r) * 2 + 1];
    v2f ax;
    ax.x = hi ? (x0 * x0 + x1 * x1) : x0;
    ax.y = hi ? 1.0f                : x1;

    const float y0 = sig[(m0 + r) * 2 + 0];
    const float y1 = sig[(m0 + r) * 2 + 1];
    v2f ad;
    ad.x = hi ? (y0 * y0 + y1 * y1) : y0;
    ad.y = hi ? 1.0f                : y1;

    const v8f zero = {};

#pragma unroll
    for (int tt = 0; tt < 4; ++tt) {
        const int n0 = nBase + (tt << 4);

        // ---- B matrix (4x16) from centres, SCL folded in.
        const float c0 = cen[(n0 + r) * 2 + 0];
        const float c1 = cen[(n0 + r) * 2 + 1];
        v2f b;
        b.x = hi ? SCL                         : N2SCL * c0;   // K=0 / K=2
        b.y = hi ? SCL * (c0 * c0 + c1 * c1)   : N2SCL * c1;   // K=1 / K=3

        // D = A x B + 0  ->  SCL * d2 tiles
        v8f ex = __builtin_amdgcn_wmma_f32_16x16x4_f32(
            false, ax, false, b, (short)0, zero, false, false);
        v8f ed = __builtin_amdgcn_wmma_f32_16x16x4_f32(
            false, ad, false, b, (short)0, zero, false, false);

        // C/D layout: VGPR i, lanes 0-15 -> (M=i, N=lane); 16-31 -> (M=i+8, N=lane-16)
        float* __restrict__ o =
            out + (size_t)(m0 + (hi ? 8 : 0)) * (size_t)M_DIM + (size_t)(n0 + r);

#pragma unroll
        for (int i = 0; i < 8; ++i) {
            o[(size_t)i * M_DIM] = __builtin_amdgcn_exp2f(ex[i])
                                 + __builtin_amdgcn_exp2f(ed[i]);
        }
    }
}

extern "C" void kernel_launch(void* const* d_in, const int* in_sizes, int n_in,
                              void* d_out, int out_size, void* d_ws, size_t ws_size,
                              hipStream_t stream)
{
    const float* sig = (const float*)d_in[0];   // (4196, 2) float32
    const float* cen = (const float*)d_in[1];   // (16384, 2) float32
    float* out = (float*)d_out;                 // (4096, 16384) float32

    // 8 waves per block, each wave covers 64 columns (4 N-tiles) x 16 rows.
    // grid.x = 16384 / (64 * 8) = 32 ; grid.y = 4096 / 16 = 256.
    dim3 grid(M_DIM / (64 * 8), T_DIM / 16);
    dim3 block(256);
    rbf_wmma_kernel<<<grid, block, 0, stream>>>(sig, cen, out);
}